// PointpairAttentionLayer_20100446945959
// MI455X (gfx1250) — compile-verified
//
#include <hip/hip_runtime.h>

typedef __attribute__((ext_vector_type(2))) float v2f;
typedef __attribute__((ext_vector_type(4))) float f4;
typedef __attribute__((ext_vector_type(8))) float v8f;

#define NCLS      8
#define CF        64      // IN_FEATURES
#define NPTS      4096
#define KN        20      // neighbors
#define NEG_SLOPE 0.2f

#define WSTRIDE   68      // LDS row stride for W (16B aligned, bank-skewed)
#define XSTRIDE   36      // LDS row stride for X (16B aligned, bank-skewed)
#define WAVES     4

__global__ __launch_bounds__(WAVES * 32)
void PointpairAttention_gfx1250_kernel(const float* __restrict__ x,
                                       const int*   __restrict__ core_types,
                                       const int*   __restrict__ target_types,
                                       const float* __restrict__ W,
                                       const float* __restrict__ a_pair,
                                       const float* __restrict__ a_self,
                                       float*       __restrict__ out)
{
    __shared__ float Wlds[CF * WSTRIDE];              // 17408 B
    __shared__ float Xlds[WAVES][CF * XSTRIDE];       // 36864 B

    const int tid  = threadIdx.x;
    const int wave = tid >> 5;
    const int lane = tid & 31;
    const int half = lane >> 4;     // 0: lanes 0-15, 1: lanes 16-31
    const int l16  = lane & 15;

    // CDNA5 prefetch of the hot attention tables (global_prefetch_b8)
    __builtin_prefetch(a_pair + (size_t)lane * CF, 0, 3);
    __builtin_prefetch(a_self + (size_t)(lane & 7) * CF, 0, 3);

    // ---- cooperative load of W (64x64, row-major c->d) into LDS ----
    for (int i = 0; i < 8; ++i) {
        int idx4 = tid + i * (WAVES * 32);   // 1024 float4 total
        int row  = idx4 >> 4;                // 16 float4 per row
        int col  = (idx4 & 15) << 2;
        f4 v = *(const f4*)(W + row * CF + col);
        *(f4*)(&Wlds[row * WSTRIDE + col]) = v;
    }

    // ---- which (b, n) this wave owns ----
    const int idx = blockIdx.x * WAVES + wave;   // 0 .. 32767
    const int b   = idx >> 12;
    const int n   = idx & (NPTS - 1);

    // ---- stage X tile: 64 c-rows x 20 k (zero-padded to 32) into LDS ----
    float* Xw = &Xlds[wave][0];
    #pragma unroll
    for (int h = 0; h < 2; ++h) {
        int c = lane + h * 32;
        const float* gp = x + ((size_t)(b * CF + c) * NPTS + n) * KN;  // 80B-mult base
        float* lp = &Xw[c * XSTRIDE];
        #pragma unroll
        for (int q = 0; q < 5; ++q)
            *(f4*)(lp + q * 4) = *(const f4*)(gp + q * 4);
        f4 z = {0.f, 0.f, 0.f, 0.f};
        *(f4*)(lp + 20) = z;   // zero-pad rows k = 20..31 for the padded M-tile
        *(f4*)(lp + 24) = z;
        *(f4*)(lp + 28) = z;
    }

    __syncthreads();

    // ---- GEMM: Wh(32x64) = Xpad(32x64) * W(64x64) via v_wmma_f32_16x16x4_f32 ----
    // A 16x4 layout: lane M = l&15; vgpr0/1 = K = (2*half), (2*half)+1
    // B 4x16 layout: lane N = l&15; vgpr0/1 = K rows (2*half), (2*half)+1
    // C 16x16 layout: lane N = l&15; vgpr j = M = j + 8*half
    v8f acc[2][4];
    {
        v8f z8 = {0.f,0.f,0.f,0.f,0.f,0.f,0.f,0.f};
        #pragma unroll
        for (int m = 0; m < 2; ++m)
            #pragma unroll
            for (int t = 0; t < 4; ++t)
                acc[m][t] = z8;
    }

    const int rA = 2 * half;
    #pragma unroll
    for (int kk = 0; kk < 16; ++kk) {
        const int c0 = kk * 4;
        v2f a0, a1;
        a0.x = Xw[(c0 + rA    ) * XSTRIDE +      l16];
        a0.y = Xw[(c0 + rA + 1) * XSTRIDE +      l16];
        a1.x = Xw[(c0 + rA    ) * XSTRIDE + 16 + l16];
        a1.y = Xw[(c0 + rA + 1) * XSTRIDE + 16 + l16];
        #pragma unroll
        for (int t = 0; t < 4; ++t) {
            v2f bf;
            bf.x = Wlds[(c0 + rA    ) * WSTRIDE + t * 16 + l16];
            bf.y = Wlds[(c0 + rA + 1) * WSTRIDE + t * 16 + l16];
            acc[0][t] = __builtin_amdgcn_wmma_f32_16x16x4_f32(
                false, a0, false, bf, (short)0, acc[0][t], false, false);
            acc[1][t] = __builtin_amdgcn_wmma_f32_16x16x4_f32(
                false, a1, false, bf, (short)0, acc[1][t], false, false);
        }
    }

    // ---- epilogue: gather a, leaky-relu, softmax over k (reg-resident), elu ----
    // lane half 0 owns k = {0..7} U {16..19} (12 values); half 1 owns k = {8..15}.
    const int nj   = half ? 8 : 12;
    const int base = b * NPTS + n;
    const int core = core_types[base];
    const int* tt  = target_types + (size_t)base * KN;

    const float* arow[12];
    int kidx[12];
    #pragma unroll
    for (int j = 0; j < 12; ++j) {
        int k = half ? (8 + (j & 7)) : ((j < 8) ? j : 16 + (j - 8));
        kidx[j] = k;
        if (!half && j == 0) {
            arow[j] = a_self + (size_t)core * CF;            // k == 0 -> self
        } else {
            int t  = tt[k];
            int mn = min(core, t), mx = max(core, t);
            int p  = mn * NCLS - (mn * (mn - 1)) / 2 + (mx - mn);
            arow[j] = a_pair + (size_t)p * CF;
        }
    }

    #pragma unroll
    for (int t = 0; t < 4; ++t) {
        const int d = t * 16 + l16;

        float wh[12], e[12];
        #pragma unroll
        for (int j = 0; j < 12; ++j) {
            float v;
            if (j < 8) v = acc[0][t][j];
            else       v = acc[1][t][j - 8];
            wh[j] = v;
        }

        float mx = -3.4e38f;
        #pragma unroll
        for (int j = 0; j < 12; ++j) {
            float v = wh[j] * arow[j][d];                    // coalesced gather
            v = (v >= 0.f) ? v : v * NEG_SLOPE;              // leaky_relu
            e[j] = v;
            if (j < nj) mx = fmaxf(mx, v);
        }
        mx = fmaxf(mx, __shfl_xor(mx, 16, 32));              // lane-pair reduce

        float s = 0.f;
        #pragma unroll
        for (int j = 0; j < 12; ++j) {
            float p = __expf(e[j] - mx);
            e[j] = p;
            if (j < nj) s += p;
        }
        s += __shfl_xor(s, 16, 32);
        const float inv = 1.f / s;

        float* op = out + ((size_t)(b * CF + d) * NPTS + n) * KN;
        #pragma unroll
        for (int j = 0; j < 12; ++j) {
            if (j < nj) {
                float hp = (e[j] * inv) * wh[j];             // attention * Wh
                float o  = (hp > 0.f) ? hp : (__expf(hp) - 1.f);  // elu
                op[kidx[j]] = o;
            }
        }
    }
}

extern "C" void kernel_launch(void* const* d_in, const int* in_sizes, int n_in,
                              void* d_out, int out_size, void* d_ws, size_t ws_size,
                              hipStream_t stream) {
    (void)in_sizes; (void)n_in; (void)out_size; (void)d_ws; (void)ws_size;
    const float* x    = (const float*)d_in[0];
    const int*   core = (const int*)  d_in[1];
    const int*   tt   = (const int*)  d_in[2];
    const float* W    = (const float*)d_in[3];
    const float* ap   = (const float*)d_in[4];
    const float* as   = (const float*)d_in[5];
    float*       out  = (float*)d_out;

    const int total_pts = 8 * NPTS;                  // B * N = 32768 waves
    dim3 grid(total_pts / WAVES), block(WAVES * 32);
    PointpairAttention_gfx1250_kernel<<<grid, block, 0, stream>>>(
        x, core, tt, W, ap, as, out);
}